// CNN2_53755810676888
// MI455X (gfx1250) — compile-verified
//
#include <hip/hip_runtime.h>

typedef __attribute__((ext_vector_type(2))) float v2f;
typedef __attribute__((ext_vector_type(8))) float v8f;

#define TLEN 2048
#define ROW 30
#define PADR 32
#define STRIDE 2060            // 2048 data + 12 zero cols (unroll-2 tail); 8240B pitch -> 16 distinct banks
#define NLAGS 2048             // L = 0..2047 (mirror handled analytically)
#define WG_MAIN 128
#define WAVES_PER_WG 8
#define THREADS_MAIN (WAVES_PER_WG * 32)

// workspace layout (in floats)
#define WS_Y     0                        // 30*2048
#define WS_D0    (ROW * TLEN)             // 1024 floats: S_0 as padded 32x32
#define WS_PART  (WS_D0 + 1024)           // WG_MAIN * 2048 floats: per-WG {P,Q}

#define WMMA_F32(A, B, C) \
    __builtin_amdgcn_wmma_f32_16x16x4_f32(false, (A), false, (B), (short)0, (C), false, false)

// ---------------------------------------------------------------- kernel 1: Y = W @ x
__global__ void cnn2_y_kernel(const float* __restrict__ x, const float* __restrict__ W,
                              float* __restrict__ Y) {
    int idx = blockIdx.x * blockDim.x + threadIdx.x;
    if (idx >= ROW * TLEN) return;
    int i = idx / TLEN, t = idx % TLEN;
    float acc = 0.f;
#pragma unroll
    for (int k = 0; k < ROW; ++k) acc += W[i * ROW + k] * x[k * TLEN + t];
    Y[idx] = acc;
}

// ---------------------------------------------------------------- kernel 2: lag correlations via WMMA
__launch_bounds__(THREADS_MAIN)
__global__ void cnn2_corr_kernel(const float* __restrict__ Yg, float* __restrict__ ws) {
    extern __shared__ float lds[];
    float* Yp = lds;                       // PADR * STRIDE
    float* PQ = lds + PADR * STRIDE;       // 2048: P[0..1023], Q[1024..2047] (padded 32x32 each)
    const int tid = threadIdx.x;

    for (int i = tid; i < PADR * STRIDE; i += THREADS_MAIN) Yp[i] = 0.f;
    for (int i = tid; i < 2048; i += THREADS_MAIN) PQ[i] = 0.f;
    __syncthreads();
    for (int i = tid; i < ROW * TLEN; i += THREADS_MAIN) {
        int r = i / TLEN, c = i % TLEN;
        Yp[r * STRIDE + c] = Yg[i];
    }
    __syncthreads();

    const int lane = tid & 31;
    const int wave = tid >> 5;
    const int gw   = blockIdx.x * WAVES_PER_WG + wave;   // 0..1023; lags {gw, 2047-gw}
    const int m    = lane & 15;
    const int hi   = lane >> 4;                          // 0/1: K-half select per ISA A/B layout
    const float* rowLo = Yp + m * STRIDE;                // rows 0..15
    const float* rowHi = Yp + (m + 16) * STRIDE;         // rows 16..31

    v8f P00 = {}, P01 = {}, P10 = {}, P11 = {};
    v8f Q00 = {}, Q01 = {}, Q10 = {}, Q11 = {};

    for (int which = 0; which < 2; ++which) {
        // readfirstlane: lag is wave-uniform -> scalar loop (s_cmp/s_cbranch), EXEC stays all-ones
        const int L     = __builtin_amdgcn_readfirstlane((which == 0) ? gw : (NLAGS - 1 - gw));
        const int bound = TLEN - L;
        // dual accumulators per tile: 8 independent WMMA dependency chains
        v8f ca00 = {}, ca01 = {}, ca10 = {}, ca11 = {};
        v8f cb00 = {}, cb01 = {}, cb10 = {}, cb11 = {};
        const int tb0 = 2 * hi;
        for (int t0 = 0; t0 < bound; t0 += 8) {
            {   // K block 0
                const int tb = t0 + tb0;
                const int ta = tb + L;
                v2f A0 = { rowLo[ta], rowLo[ta + 1] };
                v2f A1 = { rowHi[ta], rowHi[ta + 1] };
                v2f B0 = { rowLo[tb], rowLo[tb + 1] };
                v2f B1 = { rowHi[tb], rowHi[tb + 1] };
                ca00 = WMMA_F32(A0, B0, ca00);
                ca01 = WMMA_F32(A0, B1, ca01);
                ca10 = WMMA_F32(A1, B0, ca10);
                ca11 = WMMA_F32(A1, B1, ca11);
            }
            {   // K block 1 (tail reads land in zero-pad cols -> contribute 0)
                const int tb = t0 + 4 + tb0;
                const int ta = tb + L;
                v2f A0 = { rowLo[ta], rowLo[ta + 1] };
                v2f A1 = { rowHi[ta], rowHi[ta + 1] };
                v2f B0 = { rowLo[tb], rowLo[tb + 1] };
                v2f B1 = { rowHi[tb], rowHi[tb + 1] };
                cb00 = WMMA_F32(A0, B0, cb00);
                cb01 = WMMA_F32(A0, B1, cb01);
                cb10 = WMMA_F32(A1, B0, cb10);
                cb11 = WMMA_F32(A1, B1, cb11);
            }
        }
        v8f c00, c01, c10, c11;
#pragma unroll
        for (int r = 0; r < 8; ++r) {
            c00[r] = ca00[r] + cb00[r];
            c01[r] = ca01[r] + cb01[r];
            c10[r] = ca10[r] + cb10[r];
            c11[r] = ca11[r] + cb11[r];
        }

        // trace: C/D layout -> diag of tile (0,0) in lanes 0-7 (vgpr=lane) and 24-31 (vgpr=lane-24);
        // same for tile (16,16). Rows 30,31 are zero so padded trace == true trace.
        float d = 0.f;
#pragma unroll
        for (int r = 0; r < 8; ++r) {
            if (lane == r)      d = c00[r] + c11[r];
            if (lane == 24 + r) d = c00[r] + c11[r];
        }
#pragma unroll
        for (int off = 16; off >= 1; off >>= 1) d += __shfl_xor(d, off, 32);
        const float inv = 1.0f / d;

        v8f s00, s01, s10, s11;
#pragma unroll
        for (int r = 0; r < 8; ++r) {
            s00[r] = c00[r] * inv; s01[r] = c01[r] * inv;
            s10[r] = c10[r] * inv; s11[r] = c11[r] * inv;
            P00[r] += s00[r]; P01[r] += s01[r]; P10[r] += s10[r]; P11[r] += s11[r];
            Q00[r] += s00[r] * s00[r]; Q01[r] += s01[r] * s01[r];
            Q10[r] += s10[r] * s10[r]; Q11[r] += s11[r] * s11[r];
        }
        if (gw == 0 && which == 0) {        // wave-uniform branch: store S_0 (mirror correction term)
            float* D0 = ws + WS_D0;
#pragma unroll
            for (int r = 0; r < 8; ++r) {
                int M = r + 8 * hi, N = m;
                D0[M * 32 + N]               = s00[r];
                D0[M * 32 + (N + 16)]        = s01[r];
                D0[(M + 16) * 32 + N]        = s10[r];
                D0[(M + 16) * 32 + (N + 16)] = s11[r];
            }
        }
    }

    // deterministic wave-serial accumulation into LDS (fixed order -> bitwise reproducible)
    for (int w = 0; w < WAVES_PER_WG; ++w) {
        if (wave == w) {
#pragma unroll
            for (int r = 0; r < 8; ++r) {
                int M = r + 8 * hi, N = m;
                PQ[M * 32 + N]                    += P00[r];
                PQ[M * 32 + N + 16]               += P01[r];
                PQ[(M + 16) * 32 + N]             += P10[r];
                PQ[(M + 16) * 32 + N + 16]        += P11[r];
                PQ[1024 + M * 32 + N]             += Q00[r];
                PQ[1024 + M * 32 + N + 16]        += Q01[r];
                PQ[1024 + (M + 16) * 32 + N]      += Q10[r];
                PQ[1024 + (M + 16) * 32 + N + 16] += Q11[r];
            }
        }
        __syncthreads();
    }
    float* part = ws + WS_PART + (size_t)blockIdx.x * 2048;
    for (int i = tid; i < 2048; i += THREADS_MAIN) part[i] = PQ[i];
}

// ---------------------------------------------------------------- kernel 3: reduce + stats + tiny CNN
__launch_bounds__(256)
__global__ void cnn2_final_kernel(const float* __restrict__ ws,
                                  const float* __restrict__ c1w, const float* __restrict__ c1b,
                                  const float* __restrict__ c2w, const float* __restrict__ c2b,
                                  const float* __restrict__ fcw, const float* __restrict__ fcb,
                                  float* __restrict__ out) {
    __shared__ float P[1024], Q[1024];
    __shared__ float Gm[900], Gv[900];
    __shared__ float h1[8 * 29 * 29];
    __shared__ float h1p[72];
    __shared__ float h2[256];
    __shared__ float h2p[16];
    const int tid = threadIdx.x;
    const float* part = ws + WS_PART;

    for (int i = tid; i < 2048; i += 256) {
        float s = 0.f;
        for (int w = 0; w < WG_MAIN; ++w) s += part[(size_t)w * 2048 + i];
        if (i < 1024) P[i] = s; else Q[i - 1024] = s;
    }
    __syncthreads();

    const float* D0 = ws + WS_D0;
    for (int e = tid; e < 900; e += 256) {
        int i = e / 30, j = e % 30;
        float d  = D0[j * 32 + i];
        float sm = (P[i * 32 + j] + P[j * 32 + i] - d) * (1.0f / 4095.0f);
        float sq = Q[i * 32 + j] + Q[j * 32 + i] - d * d;
        float sv = (sq - 4095.0f * sm * sm) * (1.0f / 4094.0f);   // ddof=1
        Gm[e] = 0.5f * (sm + 1.0f);
        Gv[e] = 0.5f * (sv + 1.0f);
    }
    __syncthreads();

    // conv1 (8,2,4,4), pad 1 -> (8,29,29), leaky
    for (int o = tid; o < 8 * 29 * 29; o += 256) {
        int c = o / 841, rem = o % 841, oy = rem / 29, ox = rem % 29;
        float acc = c1b[c];
        for (int ic = 0; ic < 2; ++ic) {
            const float* g = (ic == 0) ? Gm : Gv;
            const float* wp = c1w + (c * 2 + ic) * 16;
#pragma unroll
            for (int ky = 0; ky < 4; ++ky) {
                int yy = oy + ky - 1;
                if (yy < 0 || yy >= 30) continue;
#pragma unroll
                for (int kx = 0; kx < 4; ++kx) {
                    int xx = ox + kx - 1;
                    if (xx < 0 || xx >= 30) continue;
                    acc += g[yy * 30 + xx] * wp[ky * 4 + kx];
                }
            }
        }
        h1[o] = (acc > 0.f) ? acc : 0.2f * acc;
    }
    __syncthreads();

    // maxpool 8x8 VALID -> (8,3,3)
    for (int o = tid; o < 72; o += 256) {
        int c = o / 9, rem = o % 9, py = rem / 3, px = rem % 3;
        float mx = -3.402823466e38f;
        for (int dy = 0; dy < 8; ++dy)
            for (int dx = 0; dx < 8; ++dx)
                mx = fmaxf(mx, h1[c * 841 + (py * 8 + dy) * 29 + (px * 8 + dx)]);
        h1p[o] = mx;
    }
    __syncthreads();

    // conv2 (16,8,2,2), pad 1 -> (16,4,4), leaky
    for (int o = tid; o < 256; o += 256) {
        int c = o / 16, rem = o % 16, oy = rem / 4, ox = rem % 4;
        float acc = c2b[c];
        for (int ic = 0; ic < 8; ++ic)
#pragma unroll
            for (int ky = 0; ky < 2; ++ky) {
                int yy = oy + ky - 1;
                if (yy < 0 || yy >= 3) continue;
#pragma unroll
                for (int kx = 0; kx < 2; ++kx) {
                    int xx = ox + kx - 1;
                    if (xx < 0 || xx >= 3) continue;
                    acc += h1p[ic * 9 + yy * 3 + xx] * c2w[((c * 8 + ic) * 2 + ky) * 2 + kx];
                }
            }
        h2[o] = (acc > 0.f) ? acc : 0.2f * acc;
    }
    __syncthreads();

    // maxpool 4x4 -> (16,)
    for (int o = tid; o < 16; o += 256) {
        float mx = -3.402823466e38f;
        for (int k = 0; k < 16; ++k) mx = fmaxf(mx, h2[o * 16 + k]);
        h2p[o] = mx;
    }
    __syncthreads();

    if (tid < 2) {
        float acc = fcb[tid];
        for (int c = 0; c < 16; ++c) acc += h2p[c] * fcw[tid * 16 + c];
        out[tid] = acc;
    }
}

// ---------------------------------------------------------------- launch
extern "C" void kernel_launch(void* const* d_in, const int* in_sizes, int n_in,
                              void* d_out, int out_size, void* d_ws, size_t ws_size,
                              hipStream_t stream) {
    const float* x   = (const float*)d_in[0];
    const float* W   = (const float*)d_in[1];
    const float* c1w = (const float*)d_in[2];
    const float* c1b = (const float*)d_in[3];
    const float* c2w = (const float*)d_in[4];
    const float* c2b = (const float*)d_in[5];
    const float* fcw = (const float*)d_in[6];
    const float* fcb = (const float*)d_in[7];
    float* ws  = (float*)d_ws;
    float* out = (float*)d_out;

    cnn2_y_kernel<<<(ROW * TLEN + 255) / 256, 256, 0, stream>>>(x, W, ws + WS_Y);

    size_t ldsBytes = (size_t)(PADR * STRIDE + 2048) * sizeof(float);   // ~272 KB < 320 KB/WGP
    cnn2_corr_kernel<<<WG_MAIN, THREADS_MAIN, ldsBytes, stream>>>(ws + WS_Y, ws);

    cnn2_final_kernel<<<1, 256, 0, stream>>>(ws, c1w, c1b, c2w, c2b, fcw, fcb, out);
}